// ShiftMSE_15204184228321
// MI455X (gfx1250) — compile-verified
//
#include <hip/hip_runtime.h>
#include <hip/hip_bf16.h>

#define B_DIM   128
#define T_DIM   32768
#define NSHIFT  11               // shifts -5..+5
#define TILE    1024             // floats of y per LDS tile (per buffer)
#define HALO    16               // halo floats each side (need >=8)
#define YBUF_F  (TILE + 2*HALO)  // 1056 floats = 4224 bytes
#define NTILES  (T_DIM / TILE)   // 32
#define SPLIT   4                // segments per row -> 512 workgroups
#define SEGTILES (NTILES / SPLIT) // 8 tiles per segment
#define NTHR    256              // 8 waves (wave32)
#define EXTRA   (YBUF_F/4 - NTHR) // 8 extra float4 loads for the halo

// ---- CDNA5 async-tensor path: DMA global -> LDS, tracked by ASYNCcnt ----
__device__ __forceinline__ void async_load_f4(unsigned lds_byte,
                                              unsigned goff_byte,
                                              const float* sbase) {
  // GVS mode: mem = SGPR64 + VGPR32 byte offset; LDS dest = VGPR lds offset.
  asm volatile("global_load_async_to_lds_b128 %0, %1, %2"
               :: "v"(lds_byte), "v"(goff_byte), "s"(sbase)
               : "memory");
}
__device__ __forceinline__ void wait_async0() {
  asm volatile("s_wait_asynccnt 0x0" ::: "memory");
}

// Kernel 1: each block = one (row, segment); writes 11 partial ssq sums.
__global__ __launch_bounds__(NTHR)
void shiftmse_partial(const float* __restrict__ x,
                      const float* __restrict__ y,
                      float* __restrict__ partial) {
  const int blk = blockIdx.x;
  const int row = blk / SPLIT;
  const int seg = blk % SPLIT;
  const int tile0 = seg * SEGTILES;
  const int tid = threadIdx.x;
  const float* __restrict__ xrow = x + (size_t)row * T_DIM;
  const float* __restrict__ yrow = y + (size_t)row * T_DIM;

  __shared__ __align__(16) float ybuf[2][YBUF_F];
  __shared__ float red[NTHR / 32][NSHIFT];

  float acc[NSHIFT];
#pragma unroll
  for (int s = 0; s < NSHIFT; ++s) acc[s] = 0.0f;

  // Low 32 bits of the flat shared address == LDS byte offset on CDNA5
  // (aperture lives in the high dword), which is what async-to-LDS VDST takes.
  const unsigned ldsA = (unsigned)(size_t)(void*)&ybuf[0][0];
  const unsigned ldsB = (unsigned)(size_t)(void*)&ybuf[1][0];

  auto issue_tile = [&](int tile, unsigned ldsbase) {
    // window covers y[row, tile*TILE - HALO .. tile*TILE + TILE + HALO)
    int gf = tile * TILE - HALO + tid * 4;            // float index in row
    gf = gf < 0 ? 0 : gf;                             // clamp to row bounds
    gf = gf > (T_DIM - 4) ? (T_DIM - 4) : gf;         // (junk is masked later)
    async_load_f4(ldsbase + (unsigned)tid * 16u, (unsigned)(gf * 4), yrow);
    if (tid < EXTRA) {
      int gf2 = tile * TILE - HALO + (NTHR + tid) * 4;
      gf2 = gf2 < 0 ? 0 : gf2;
      gf2 = gf2 > (T_DIM - 4) ? (T_DIM - 4) : gf2;
      async_load_f4(ldsbase + (unsigned)(NTHR + tid) * 16u,
                    (unsigned)(gf2 * 4), yrow);
    }
  };

  issue_tile(tile0, ldsA);
  wait_async0();
  __syncthreads();

  for (int lt = 0; lt < SEGTILES; ++lt) {
    const int tile = tile0 + lt;
    // Prefetch next tile into the other buffer (overlaps compute below).
    if (lt + 1 < SEGTILES) issue_tile(tile + 1, (lt & 1) ? ldsA : ldsB);

    const float* yb = ybuf[lt & 1];
    // 20-float window: y[row, tile*TILE + 4*tid - 8 .. +11], 5x ds_load_b128
    float w[20];
#pragma unroll
    for (int k = 0; k < 5; ++k) {
      float4 v = reinterpret_cast<const float4*>(yb)[tid + 2 + k];
      w[4 * k + 0] = v.x; w[4 * k + 1] = v.y;
      w[4 * k + 2] = v.z; w[4 * k + 3] = v.w;
    }
    float4 xv = reinterpret_cast<const float4*>(xrow + tile * TILE)[tid];
    float xj[4] = {xv.x, xv.y, xv.z, xv.w};

    if (tile == 0 || tile == NTILES - 1) {       // wave-uniform edge path
      const int tbase = tile * TILE + 4 * tid;
#pragma unroll
      for (int j = 0; j < 4; ++j) {
#pragma unroll
        for (int s = -5; s <= 5; ++s) {
          float d = xj[j] - w[j + s + 8];
          int tg = tbase + j + s;
          d = ((unsigned)tg < (unsigned)T_DIM) ? d : 0.0f;
          acc[s + 5] = fmaf(d, d, acc[s + 5]);
        }
      }
    } else {                                     // hot path: no masking
#pragma unroll
      for (int j = 0; j < 4; ++j) {
#pragma unroll
        for (int s = -5; s <= 5; ++s) {
          float d = xj[j] - w[j + s + 8];
          acc[s + 5] = fmaf(d, d, acc[s + 5]);
        }
      }
    }

    wait_async0();       // next tile resident
    __syncthreads();     // all waves' async writes visible
  }

  // ---- block reduction: 11 sums over 256 threads (wave32 shuffles) ----
  const int lane = tid & 31, wid = tid >> 5;
#pragma unroll
  for (int s = 0; s < NSHIFT; ++s) {
    float v = acc[s];
    for (int off = 16; off > 0; off >>= 1) v += __shfl_down(v, off, 32);
    if (lane == 0) red[wid][s] = v;
  }
  __syncthreads();
  if (tid < NSHIFT) {
    float ssq = 0.0f;
#pragma unroll
    for (int w8 = 0; w8 < NTHR / 32; ++w8) ssq += red[w8][tid];
    partial[(size_t)blk * NSHIFT + tid] = ssq;
  }
}

// Kernel 2: one block, one thread per row -> combine segments, metric,
// min over shifts, mean over batch.
__global__ __launch_bounds__(B_DIM)
void combine_mean(const float* __restrict__ partial, float* __restrict__ out) {
  __shared__ float sred[B_DIM / 32];
  const int r = threadIdx.x;   // row

  float ssq[NSHIFT];
#pragma unroll
  for (int s = 0; s < NSHIFT; ++s) ssq[s] = 0.0f;
#pragma unroll
  for (int seg = 0; seg < SPLIT; ++seg) {
    const float* p = partial + ((size_t)r * SPLIT + seg) * NSHIFT;
#pragma unroll
    for (int s = 0; s < NSHIFT; ++s) ssq[s] += p[s];
  }

  float m = 3.4e38f;
#pragma unroll
  for (int s = 0; s < NSHIFT; ++s) {
    int sh = s - 5;
    float n = (float)(T_DIM - (sh < 0 ? -sh : sh));
    float v = fminf(ssq[s] / n, sqrtf(ssq[s]));
    m = fminf(m, v);
  }

  // mean over 128 rows
  for (int off = 16; off > 0; off >>= 1) m += __shfl_down(m, off, 32);
  if ((r & 31) == 0) sred[r >> 5] = m;
  __syncthreads();
  if (r == 0) {
    float sum = 0.0f;
#pragma unroll
    for (int i = 0; i < B_DIM / 32; ++i) sum += sred[i];
    out[0] = sum * (1.0f / (float)B_DIM);
  }
}

extern "C" void kernel_launch(void* const* d_in, const int* in_sizes, int n_in,
                              void* d_out, int out_size, void* d_ws, size_t ws_size,
                              hipStream_t stream) {
  const float* x = (const float*)d_in[0];
  const float* y = (const float*)d_in[1];
  float* partial = (float*)d_ws;            // 512*11 floats = 22.5 KB scratch
  float* out     = (float*)d_out;           // scalar fp32 result
  (void)in_sizes; (void)n_in; (void)out_size; (void)ws_size;

  shiftmse_partial<<<B_DIM * SPLIT, NTHR, 0, stream>>>(x, y, partial);
  combine_mean<<<1, B_DIM, 0, stream>>>(partial, out);
}